// TFTransfoXLMainLayer_35613868819142
// MI455X (gfx1250) — compile-verified
//
#include <hip/hip_runtime.h>

// ---------------------------------------------------------------------------
// Transformer-XL forward for MI455X (gfx1250), fp32 end-to-end.
// All GEMMs via V_WMMA_F32_16X16X4_F32 (wave32, 16x16 tile, K-step 4).
// Attention is fused: scores live in LDS (never hit HBM), eliminating the
// dominant ~768MB/layer of score traffic (HBM roofline: 23.3 TB/s).
// GEMM wave tile is 32x64: 2 A-fragments share 4 B-fragments -> 8 WMMA per
// K-step, halving B-load instruction pressure vs a 16x64 tile.
// ---------------------------------------------------------------------------

typedef __attribute__((ext_vector_type(2))) float v2f;
typedef __attribute__((ext_vector_type(8))) float v8f;

#define D_MODEL 1024
#define N_HEAD  16
#define D_HEAD  64
#define HDIM    1024
#define D_INNER 4096
#define QLEN    512
#define MLEN    512
#define KLEN    1024
#define BSZ     4
#define NLAYER  4

static __device__ __forceinline__ v8f wmma4(v2f a, v2f b, v8f c) {
  return __builtin_amdgcn_wmma_f32_16x16x4_f32(false, a, false, b, (short)0, c,
                                               false, false);
}

// ---------------------------------------------------------------------------
// Generic strided fp32 WMMA GEMM:  C[m,n] = sum_k A[m,k]*B[k,n] (+bias, ReLU)
// A element (m,k) at A + m*sAm + k          (A is always K-contiguous here)
// B element (k,n) at B + k*sBk + n*sBn      (BK1: sBk==1; else sBn==1)
// C element (m,n) at C + m*sCm + n
// Wave tile: 32(M) x 64(N); 4 waves/block -> 128x64 block tile.
// Fragments (wave32):
//   A 16x4 : lanes 0-15 rows M=0..15 {K=k,k+1}; lanes 16-31 {K=k+2,k+3}
//   B 4x16 : same K->(half,vgpr) map, N striped on lanes 0..15
//   C 16x16: vgpr v -> row v (lanes<16) / v+8 (lanes>=16), N = lane&15
// ---------------------------------------------------------------------------
template<bool BK1>
__global__ __launch_bounds__(128) void gemm_wmma_f32(
    const float* __restrict__ A, const float* __restrict__ B,
    float* __restrict__ C,
    int M, int N, int K,
    int sAm, int sBk, long long sBn, int sCm,
    const float* __restrict__ bias, int relu)
{
  (void)M; (void)N;
  const int lane = threadIdx.x & 31;
  const int wv   = threadIdx.x >> 5;
  const int half = lane >> 4;
  const int r    = lane & 15;
  const int m0   = blockIdx.y * 128 + wv * 32;
  const int n0   = blockIdx.x * 64;
  const int c0   = n0 + r;

  const float* Ar0 = A + (long long)(m0 + r) * sAm;
  const float* Ar1 = A + (long long)(m0 + 16 + r) * sAm;

  v8f c00 = {}, c01 = {}, c02 = {}, c03 = {};
  v8f c10 = {}, c11 = {}, c12 = {}, c13 = {};

#pragma unroll 2
  for (int k = 0; k < K; k += 4) {
    const int ka = k + (half << 1);
    v2f a0 = *(const v2f*)(Ar0 + ka);
    v2f a1 = *(const v2f*)(Ar1 + ka);
    v2f b0, b1, b2, b3;
    if (BK1) {
      const float* Bk = B + ka;
      b0 = *(const v2f*)(Bk + (long long)(c0      ) * sBn);
      b1 = *(const v2f*)(Bk + (long long)(c0 + 16 ) * sBn);
      b2 = *(const v2f*)(Bk + (long long)(c0 + 32 ) * sBn);
      b3 = *(const v2f*)(Bk + (long long)(c0 + 48 ) * sBn);
    } else {
      const float* Bk0 = B + (long long)ka * sBk;
      const float* Bk1 = Bk0 + sBk;
      b0.x = Bk0[c0     ]; b0.y = Bk1[c0     ];
      b1.x = Bk0[c0 + 16]; b1.y = Bk1[c0 + 16];
      b2.x = Bk0[c0 + 32]; b2.y = Bk1[c0 + 32];
      b3.x = Bk0[c0 + 48]; b3.y = Bk1[c0 + 48];
    }
    c00 = wmma4(a0, b0, c00);  c10 = wmma4(a1, b0, c10);
    c01 = wmma4(a0, b1, c01);  c11 = wmma4(a1, b1, c11);
    c02 = wmma4(a0, b2, c02);  c12 = wmma4(a1, b2, c12);
    c03 = wmma4(a0, b3, c03);  c13 = wmma4(a1, b3, c13);
  }

#pragma unroll
  for (int g = 0; g < 2; g++) {
    const int rb = m0 + g * 16 + (half << 3);
#pragma unroll
    for (int t = 0; t < 4; t++) {
      const int col = c0 + t * 16;
      const float bs = bias ? bias[col] : 0.0f;
      v8f acc;
      if (g == 0) acc = (t == 0) ? c00 : (t == 1) ? c01 : (t == 2) ? c02 : c03;
      else        acc = (t == 0) ? c10 : (t == 1) ? c11 : (t == 2) ? c12 : c13;
#pragma unroll
      for (int v = 0; v < 8; v++) {
        float x = acc[v] + bs;
        if (relu) x = fmaxf(x, 0.0f);
        C[(long long)(rb + v) * sCm + col] = x;
      }
    }
  }
}

// ---------------------------------------------------------------------------
// Fused relative attention. One block = (b, head n, 32-query strip).
// LDS holds the 32x1024 score strip (padded rows -> conflict-free phase-4
// A-fragment ds_load_b64). Phases:
//  1) S  = q_rw(32x64) @ K^T(64x1024)          [WMMA -> LDS]
//  2) S += rel_shift( q_rr(32x64) @ R^T )      [WMMA, fragment scatter-add]
//     rel_shift: BD[i,j] = pre[i, m] with j = m + i - (QLEN-1)
//  3) masked softmax rows in LDS (mask j-i>MLEN), store exp + 1/sum
//  4) O = P(32x1024) @ V(1024x64), scale by 1/sum -> avec
// ---------------------------------------------------------------------------
#define SROW 1028   // 1024 + 4 pad: phase-4 lane bank stride = 4 -> no conflicts

__global__ __launch_bounds__(256) void attn_fused(
    const float* __restrict__ qrw, const float* __restrict__ qrr,
    const float* __restrict__ wh, const float* __restrict__ rh,
    float* __restrict__ avec)
{
  const int ib = blockIdx.x;        // query strip (16 strips of 32)
  const int n  = blockIdx.y;        // head
  const int b  = blockIdx.z;        // batch
  const int i0 = ib * 32;

  __shared__ float S[32 * SROW];
  __shared__ float sinv[32];

  const int tid  = threadIdx.x;
  const int lane = tid & 31;
  const int wv   = tid >> 5;        // 0..7
  const int half = lane >> 4;
  const int r    = lane & 15;

  const float* Kb = wh + HDIM     + (long long)b * 3072 + n * 64; // (kk,j): +j*12288+kk
  const float* Vb = wh + 2 * HDIM + (long long)b * 3072 + n * 64; // (j,d) : +j*12288+d
  const float* Rb = rh + n * 64;                                  // (kk,m): +m*1024+kk

  const int mt = (wv & 1) * 16;     // local m-tile (rows)
  const int nq = (wv >> 1) * 256;   // 256-wide column strip

  // ---------------- Phase 1: S = q_rw @ K^T ----------------
  {
    const float* Aw = qrw + ((long long)(i0 + mt + r) * 4 + b) * 1024 + n * 64;
    for (int nt = 0; nt < 4; nt++) {
      const int c0 = nq + nt * 64 + r;
      v8f a0 = {}, a1 = {}, a2 = {}, a3 = {};
#pragma unroll 4
      for (int k = 0; k < D_HEAD; k += 4) {
        const int ka = k + (half << 1);
        v2f av = *(const v2f*)(Aw + ka);
        v2f b0 = *(const v2f*)(Kb + (long long)(c0     ) * 12288 + ka);
        v2f b1 = *(const v2f*)(Kb + (long long)(c0 + 16) * 12288 + ka);
        v2f b2 = *(const v2f*)(Kb + (long long)(c0 + 32) * 12288 + ka);
        v2f b3 = *(const v2f*)(Kb + (long long)(c0 + 48) * 12288 + ka);
        a0 = wmma4(av, b0, a0); a1 = wmma4(av, b1, a1);
        a2 = wmma4(av, b2, a2); a3 = wmma4(av, b3, a3);
      }
      const int rb = mt + (half << 3);
#pragma unroll
      for (int t = 0; t < 4; t++) {
        v8f acc = (t == 0) ? a0 : (t == 1) ? a1 : (t == 2) ? a2 : a3;
#pragma unroll
        for (int v = 0; v < 8; v++)
          S[(rb + v) * SROW + c0 + t * 16] = acc[v];
      }
    }
  }
  __syncthreads();

  // ---------------- Phase 2: S += shifted (q_rr @ R^T) ----------------
  {
    const float* Aw = qrr + ((long long)(i0 + mt + r) * 4 + b) * 1024 + n * 64;
    for (int nt = 0; nt < 4; nt++) {
      const int c0m = nq + nt * 64 + r;   // m coordinate
      v8f a0 = {}, a1 = {}, a2 = {}, a3 = {};
#pragma unroll 4
      for (int k = 0; k < D_HEAD; k += 4) {
        const int ka = k + (half << 1);
        v2f av = *(const v2f*)(Aw + ka);
        v2f b0 = *(const v2f*)(Rb + (long long)(c0m     ) * 1024 + ka);
        v2f b1 = *(const v2f*)(Rb + (long long)(c0m + 16) * 1024 + ka);
        v2f b2 = *(const v2f*)(Rb + (long long)(c0m + 32) * 1024 + ka);
        v2f b3 = *(const v2f*)(Rb + (long long)(c0m + 48) * 1024 + ka);
        a0 = wmma4(av, b0, a0); a1 = wmma4(av, b1, a1);
        a2 = wmma4(av, b2, a2); a3 = wmma4(av, b3, a3);
      }
      const int rb = mt + (half << 3);
#pragma unroll
      for (int t = 0; t < 4; t++) {
        const int m = c0m + t * 16;
        v8f acc = (t == 0) ? a0 : (t == 1) ? a1 : (t == 2) ? a2 : a3;
#pragma unroll
        for (int v = 0; v < 8; v++) {
          const int row = rb + v;
          const int jj = m + (i0 + row) - (QLEN - 1);   // rel-shifted column
          if (jj >= 0) S[row * SROW + jj] += acc[v];    // jj<=1023 guaranteed
        }
      }
    }
  }
  __syncthreads();

  // ---------------- Phase 3: masked softmax per row ----------------
  {
    for (int rr = 0; rr < 4; rr++) {
      const int row = wv * 4 + rr;
      const int ig  = i0 + row;
      float mx = -3.0e38f;
      for (int j = lane; j < KLEN; j += 32) {
        const float s = S[row * SROW + j];
        if ((j - ig) <= MLEN) mx = fmaxf(mx, s);
      }
#pragma unroll
      for (int off = 16; off > 0; off >>= 1) mx = fmaxf(mx, __shfl_xor(mx, off, 32));
      float sum = 0.0f;
      for (int j = lane; j < KLEN; j += 32) {
        const float s = S[row * SROW + j];
        const float p = ((j - ig) <= MLEN) ? __expf((s - mx) * 0.125f) : 0.0f;
        S[row * SROW + j] = p;
        sum += p;
      }
#pragma unroll
      for (int off = 16; off > 0; off >>= 1) sum += __shfl_xor(sum, off, 32);
      if (lane == 0) sinv[row] = 1.0f / sum;
    }
  }
  __syncthreads();

  // ---------------- Phase 4: O = P @ V (scaled by 1/sum) ----------------
  {
    const int mt4 = (wv >> 2) * 16;
    const int nt4 = (wv & 3) * 16;
    const float* Sr = &S[(mt4 + r) * SROW];
    v8f o = {};
#pragma unroll 4
    for (int k = 0; k < KLEN; k += 4) {
      const int ka = k + (half << 1);
      v2f av = *(const v2f*)(Sr + ka);
      const float* Bk0 = Vb + (long long)ka * 12288;
      v2f bv; bv.x = Bk0[nt4 + r]; bv.y = Bk0[12288 + nt4 + r];
      o = wmma4(av, bv, o);
    }
    const int rb4 = mt4 + (half << 3);
#pragma unroll
    for (int v = 0; v < 8; v++) {
      const int row = rb4 + v;
      avec[((long long)(i0 + row) * 4 + b) * 1024 + n * 64 + nt4 + r] =
          o[v] * sinv[row];
    }
  }
}

// ---------------------------------------------------------------------------
// Adaptive embedding: one block per token (row = q*BSZ+b of h), gathers the
// cluster row into LDS and projects d_i -> 1024, scaled by sqrt(D_MODEL)=32.
// ---------------------------------------------------------------------------
__global__ __launch_bounds__(256) void embed_kernel(
    const int* __restrict__ ids,
    const float* __restrict__ e0, const float* __restrict__ e1,
    const float* __restrict__ e2, const float* __restrict__ e3,
    const float* __restrict__ p0, const float* __restrict__ p1,
    const float* __restrict__ p2, const float* __restrict__ p3,
    float* __restrict__ h)
{
  const int blk = blockIdx.x;          // = q*4 + b
  const int q = blk >> 2, b = blk & 3;
  const int tok = ids[b * QLEN + q];

  const float* E; const float* P; int d; long long off;
  if (tok < 20000)       { E = e0; P = p0; d = 1024; off = (long long)tok * 1024; }
  else if (tok < 40000)  { E = e1; P = p1; d = 256;  off = (long long)(tok - 20000) * 256; }
  else if (tok < 200000) { E = e2; P = p2; d = 64;   off = (long long)(tok - 40000) * 64; }
  else                   { E = e3; P = p3; d = 16;   off = (long long)(tok - 200000) * 16; }

  __shared__ float erow[1024];
  for (int t = threadIdx.x; t < d; t += 256) erow[t] = E[off + t];
  __syncthreads();

#pragma unroll
  for (int s = 0; s < 4; s++) {
    const int e = threadIdx.x + (s << 8);
    float acc = 0.0f;
    for (int dd = 0; dd < d; dd++) acc += erow[dd] * P[(long long)dd * 1024 + e];
    h[(long long)blk * 1024 + e] = acc * 32.0f;   // sqrt(1024)
  }
}

// pos_emb[k][j] : pos = klen-1-k ; first 512 = sin, last 512 = cos
__global__ __launch_bounds__(256) void posemb_kernel(float* __restrict__ p)
{
  const int idx = blockIdx.x * 256 + threadIdx.x;   // 1024*1024
  const int k = idx >> 10, j = idx & 1023;
  const float pos = (float)(KLEN - 1 - k);
  const int jj = j & 511;
  const float invf = __expf(-(float)jj * (9.210340371976184f / 512.0f)); // 1e4^(-jj/512)
  const float v = pos * invf;
  p[idx] = (j < 512) ? __sinf(v) : __cosf(v);
}

// q_rw/q_rr = q_head + r_w_bias / r_r_bias ; q is rows [mlen:] of w_heads[:,:,:H]
__global__ __launch_bounds__(256) void build_q(
    const float* __restrict__ wh, const float* __restrict__ rwb,
    const float* __restrict__ rrb,
    float* __restrict__ qrw, float* __restrict__ qrr)
{
  const long long idx = (long long)blockIdx.x * 256 + threadIdx.x;   // 512*4*1024
  const int hd = (int)(idx & 1023);
  const long long ib = idx >> 10;                    // i*4+b
  const float q = wh[(2048 + ib) * 3072 + hd];       // row (512+i)*4+b, first H cols
  qrw[idx] = q + rwb[hd];
  qrr[idx] = q + rrb[hd];
}

// h = LayerNorm(h + res) * g + b ; one block per row of 1024
__global__ __launch_bounds__(256) void ln_residual(
    float* __restrict__ h, const float* __restrict__ res,
    const float* __restrict__ g, const float* __restrict__ be)
{
  const int row = blockIdx.x;
  const long long base = (long long)row * D_MODEL;
  const int tid = threadIdx.x;

  float x[4]; float s = 0.0f;
#pragma unroll
  for (int k = 0; k < 4; k++) {
    const int j = tid + (k << 8);
    x[k] = h[base + j] + res[base + j];
    s += x[k];
  }
#pragma unroll
  for (int off = 16; off > 0; off >>= 1) s += __shfl_xor(s, off, 32);
  __shared__ float r1[8], r2[8];
  if ((tid & 31) == 0) r1[tid >> 5] = s;
  __syncthreads();
  float tot = 0.0f;
#pragma unroll
  for (int w = 0; w < 8; w++) tot += r1[w];
  const float mu = tot * (1.0f / 1024.0f);

  float vs = 0.0f;
#pragma unroll
  for (int k = 0; k < 4; k++) { const float d = x[k] - mu; vs += d * d; }
#pragma unroll
  for (int off = 16; off > 0; off >>= 1) vs += __shfl_xor(vs, off, 32);
  if ((tid & 31) == 0) r2[tid >> 5] = vs;
  __syncthreads();
  float vt = 0.0f;
#pragma unroll
  for (int w = 0; w < 8; w++) vt += r2[w];
  const float inv = rsqrtf(vt * (1.0f / 1024.0f) + 1e-5f);

#pragma unroll
  for (int k = 0; k < 4; k++) {
    const int j = tid + (k << 8);
    h[base + j] = (x[k] - mu) * inv * g[j] + be[j];
  }
}

// out[b,q,:] = h[q,b,:]
__global__ __launch_bounds__(256) void tpose_kernel(
    const float* __restrict__ h, float* __restrict__ out)
{
  const long long idx = (long long)blockIdx.x * 256 + threadIdx.x;   // 2M
  const int dcol = (int)(idx & 1023);
  const long long qb = idx >> 10;
  const int q = (int)(qb >> 2), b = (int)(qb & 3);
  out[((long long)b * QLEN + q) * 1024 + dcol] = h[idx];
}

// ---------------------------------------------------------------------------
extern "C" void kernel_launch(void* const* d_in, const int* in_sizes, int n_in,
                              void* d_out, int out_size, void* d_ws, size_t ws_size,
                              hipStream_t stream)
{
  (void)in_sizes; (void)n_in; (void)out_size; (void)ws_size;

  const int*   ids  = (const int*)d_in[0];
  const float* mems = (const float*)d_in[1];
  const float* e0 = (const float*)d_in[2],  *p0 = (const float*)d_in[3];
  const float* e1 = (const float*)d_in[4],  *p1 = (const float*)d_in[5];
  const float* e2 = (const float*)d_in[6],  *p2 = (const float*)d_in[7];
  const float* e3 = (const float*)d_in[8],  *p3 = (const float*)d_in[9];
  const float* qkvW = (const float*)d_in[10];
  const float* rW   = (const float*)d_in[11];
  const float* oW   = (const float*)d_in[12];
  const float* rwb  = (const float*)d_in[13];
  const float* rrb  = (const float*)d_in[14];
  const float* ln1g = (const float*)d_in[15], *ln1b = (const float*)d_in[16];
  const float* ffw1 = (const float*)d_in[17], *ffb1 = (const float*)d_in[18];
  const float* ffw2 = (const float*)d_in[19], *ffb2 = (const float*)d_in[20];
  const float* ln2g = (const float*)d_in[21], *ln2b = (const float*)d_in[22];

  float* ws = (float*)d_ws;
  float* h    = ws; ws += 2097152;    // (512,4,1024) hidden
  float* pos  = ws; ws += 1048576;    // (1024,1024)  pos_emb
  float* cat  = ws; ws += 4194304;    // (1024,4,1024)
  float* wh   = ws; ws += 12582912;   // (1024,4,3072) w_heads
  float* rh   = ws; ws += 1048576;    // (1024,1024)  r_head_k
  float* qrw  = ws; ws += 2097152;    // (512,4,16,64)
  float* qrr  = ws; ws += 2097152;
  float* avec = ws; ws += 2097152;    // (512,4,1024)
  float* aout = ws; ws += 2097152;    // (512,4,1024)
  float* ffp  = ws; ws += 8388608;    // (2048,4096)
  float* ffo  = ws; ws += 2097152;    // (2048,1024)

  embed_kernel<<<2048, 256, 0, stream>>>(ids, e0, e1, e2, e3, p0, p1, p2, p3, h);
  posemb_kernel<<<4096, 256, 0, stream>>>(pos);

  for (int l = 0; l < NLAYER; l++) {
    // cat = [mems[l]; h]
    hipMemcpyAsync(cat, mems + (size_t)l * 2097152, 2097152 * sizeof(float),
                   hipMemcpyDeviceToDevice, stream);
    hipMemcpyAsync(cat + 2097152, h, 2097152 * sizeof(float),
                   hipMemcpyDeviceToDevice, stream);

    // w_heads = cat @ qkv_W[l]            (4096 x 3072 x 1024)
    gemm_wmma_f32<false><<<dim3(48, 32, 1), 128, 0, stream>>>(
        cat, qkvW + (size_t)l * 1024 * 3072, wh,
        4096, 3072, 1024, 1024, 3072, 1, 3072, nullptr, 0);

    // r_head_k = pos_emb @ r_W[l]         (1024 x 1024 x 1024)
    gemm_wmma_f32<false><<<dim3(16, 8, 1), 128, 0, stream>>>(
        pos, rW + (size_t)l * 1024 * 1024, rh,
        1024, 1024, 1024, 1024, 1024, 1, 1024, nullptr, 0);

    build_q<<<8192, 256, 0, stream>>>(wh, rwb + l * 1024, rrb + l * 1024, qrw, qrr);

    // fused rel-attention: scores in LDS, writes attn_vec
    attn_fused<<<dim3(16, 16, 4), 256, 0, stream>>>(qrw, qrr, wh, rh, avec);

    // attn_out = attn_vec @ o_W[l]        (2048 x 1024 x 1024)
    gemm_wmma_f32<false><<<dim3(16, 16, 1), 128, 0, stream>>>(
        avec, oW + (size_t)l * 1024 * 1024, aout,
        2048, 1024, 1024, 1024, 1024, 1, 1024, nullptr, 0);

    ln_residual<<<2048, 256, 0, stream>>>(h, aout, ln1g + l * 1024, ln1b + l * 1024);

    // ff1 = relu(h @ ff_w1 + b1)          (2048 x 4096 x 1024)
    gemm_wmma_f32<false><<<dim3(64, 16, 1), 128, 0, stream>>>(
        h, ffw1 + (size_t)l * 1024 * 4096, ffp,
        2048, 4096, 1024, 1024, 4096, 1, 4096, ffb1 + l * 4096, 1);

    // ff2 = ff1 @ ff_w2 + b2              (2048 x 1024 x 4096)
    gemm_wmma_f32<false><<<dim3(16, 16, 1), 128, 0, stream>>>(
        ffp, ffw2 + (size_t)l * 4096 * 1024, ffo,
        2048, 1024, 4096, 4096, 1024, 1, 1024, ffb2 + l * 1024, 0);

    ln_residual<<<2048, 256, 0, stream>>>(h, ffo, ln2g + l * 1024, ln2b + l * 1024);
  }

  tpose_kernel<<<8192, 256, 0, stream>>>(h, (float*)d_out);
}